// BPR_53223234732669
// MI455X (gfx1250) — compile-verified
//
#include <hip/hip_runtime.h>
#include <hip/hip_bf16.h>

#define U_NUM   50000
#define I_NUM   20000
#define F_DIM   64
#define N_EDGES 1000000
#define BATCH   131072

typedef float v2f __attribute__((ext_vector_type(2)));
typedef float v8f __attribute__((ext_vector_type(8)));

__global__ void zero_f32(float* __restrict__ p, int n) {
  int t = blockIdx.x * blockDim.x + threadIdx.x;
  if (t < n) p[t] = 0.0f;
}

// Edge-parallel SpMM: out[sidx[e], :] += vals[e] * src[gidx[e], :]
// 16 threads per edge, float4 per thread -> 256B coalesced row read,
// native global_atomic_add_f32 scatter (L2-resident tables on MI455X).
__global__ __launch_bounds__(256) void spmm_edges(
    const float* __restrict__ src, const float* __restrict__ vals,
    const int* __restrict__ gidx, const int* __restrict__ sidx,
    float* __restrict__ out) {
  int t = blockIdx.x * blockDim.x + threadIdx.x;
  int e = t >> 4;
  if (e >= N_EDGES) return;
  int f = (t & 15) << 2;
  int g = gidx[e];
  int s = sidx[e];
  float v = vals[e];
  const float4 x = *(const float4*)(src + (size_t)g * F_DIM + f);
  float* o = out + (size_t)s * F_DIM + f;
  unsafeAtomicAdd(o + 0, v * x.x);
  unsafeAtomicAdd(o + 1, v * x.y);
  unsafeAtomicAdd(o + 2, v * x.z);
  unsafeAtomicAdd(o + 3, v * x.w);
}

__global__ void combine4(const float* __restrict__ e0, const float* __restrict__ e1,
                         const float* __restrict__ e2, const float* __restrict__ e3,
                         float* __restrict__ out, int n) {
  int t = blockIdx.x * blockDim.x + threadIdx.x;
  if (t < n)
    out[t] = e0[t] + 0.5f * e1[t] + (1.0f / 3.0f) * e2[t] + 0.25f * e3[t];
}

// One wave32 per 16 batch samples. S_i = U * Pi^T and S_j = U * Pj^T via
// 16 steps of V_WMMA_F32_16X16X4_F32 each (exact f32). Diagonal of S gives
// the per-sample dot products; stable logaddexp; wave-reduce; one atomic.
__global__ __launch_bounds__(256) void contrastive_wmma(
    const int* __restrict__ idx_u, const int* __restrict__ idx_i,
    const int* __restrict__ idx_j, const float* __restrict__ degree,
    const float* __restrict__ old_emb, const float* __restrict__ gcn_emb,
    float inv_num, float* __restrict__ out) {
  int gtid = blockIdx.x * blockDim.x + threadIdx.x;
  int wave = gtid >> 5;
  int lane = threadIdx.x & 31;
  int b0   = wave * 16;
  int m    = lane & 15;       // matrix row handled by this lane
  int half = lane >> 4;       // K sub-pair selector (A/B layout for 16x16x4)

  const float* ru = old_emb + (size_t)idx_u[b0 + m] * F_DIM + 2 * half;
  const float* ri = gcn_emb + (size_t)idx_i[b0 + m] * F_DIM + 2 * half;
  const float* rj = gcn_emb + (size_t)idx_j[b0 + m] * F_DIM + 2 * half;

  v8f ci = {0.f, 0.f, 0.f, 0.f, 0.f, 0.f, 0.f, 0.f};
  v8f cj = {0.f, 0.f, 0.f, 0.f, 0.f, 0.f, 0.f, 0.f};

#pragma unroll
  for (int k = 0; k < 16; ++k) {
    v2f a  = *(const v2f*)(ru + 4 * k);
    v2f bi = *(const v2f*)(ri + 4 * k);
    v2f bj = *(const v2f*)(rj + 4 * k);
    ci = __builtin_amdgcn_wmma_f32_16x16x4_f32(false, a, false, bi, (short)0,
                                               ci, false, false);
    cj = __builtin_amdgcn_wmma_f32_16x16x4_f32(false, a, false, bj, (short)0,
                                               cj, false, false);
  }

  // C/D layout: lane L holds column N=L%16; VGPR v holds row M=v+8*(L/16).
  // Diagonal (M==N): lanes 0..7 at vgpr=lane (M=lane), lanes 24..31 at
  // vgpr=lane-24 (M=lane-16).
  float contrib = 0.0f;
  bool valid = (lane < 8) || (lane >= 24);
  if (valid) {
    int vi = (lane < 8) ? lane : (lane - 24);
    int M  = (lane < 8) ? lane : (lane - 16);
    float si = ci[vi];
    float sj = cj[vi];
    float mx = fmaxf(si, sj);
    float mn = fminf(si, sj);
    float lae = mx + log1pf(expf(mn - mx));   // logaddexp(si, sj), stable
    float lr  = si - lae;
    contrib = -lr * degree[b0 + M];
  }
#pragma unroll
  for (int off = 16; off > 0; off >>= 1)
    contrib += __shfl_xor(contrib, off, 32);
  if (lane == 0) unsafeAtomicAdd(out, contrib * inv_num);
}

extern "C" void kernel_launch(void* const* d_in, const int* in_sizes, int n_in,
                              void* d_out, int out_size, void* d_ws, size_t ws_size,
                              hipStream_t stream) {
  const float* embed_user = (const float*)d_in[0];
  const float* embed_item = (const float*)d_in[1];
  const float* old_U      = (const float*)d_in[2];
  const float* old_I      = (const float*)d_in[3];
  const int*   rows       = (const int*)d_in[4];
  const int*   cols       = (const int*)d_in[5];
  const float* ui_vals    = (const float*)d_in[6];
  const float* iu_vals    = (const float*)d_in[7];
  const int*   user       = (const int*)d_in[8];
  const int*   item_i     = (const int*)d_in[9];
  const int*   item_j     = (const int*)d_in[10];
  const float* degree_U   = (const float*)d_in[11];
  // d_in[12] item_z_U: unused by reference
  const int*   user_      = (const int*)d_in[13];
  const int*   item_i_    = (const int*)d_in[14];
  const int*   item_j_    = (const int*)d_in[15];
  const float* degree_I   = (const float*)d_in[16];
  // d_in[17] item_z_I: unused by reference

  const size_t UF = (size_t)U_NUM * F_DIM;   // 3.2M floats
  const size_t IF = (size_t)I_NUM * F_DIM;   // 1.28M floats

  float* ws    = (float*)d_ws;
  float* u1    = ws;
  float* u2    = ws + UF;
  float* u3    = ws + 2 * UF;
  float* i1    = ws + 3 * UF;
  float* i2    = i1 + IF;
  float* i3    = i1 + 2 * IF;
  float* gcn_u = i1 + 3 * IF;
  float* gcn_i = gcn_u + UF;

  // Zero the six SpMM accumulators (contiguous region) and the scalar output.
  int nz = (int)(3 * UF + 3 * IF);
  zero_f32<<<(nz + 255) / 256, 256, 0, stream>>>(ws, nz);
  zero_f32<<<1, 32, 0, stream>>>((float*)d_out, 1);

  dim3 eb(256);
  dim3 eg((N_EDGES * 16 + 255) / 256);
  // u1 = A  @ i0 ; i1 = A^T @ u0
  spmm_edges<<<eg, eb, 0, stream>>>(embed_item, ui_vals, cols, rows, u1);
  spmm_edges<<<eg, eb, 0, stream>>>(embed_user, iu_vals, rows, cols, i1);
  // u2 = A  @ i1 ; i2 = A^T @ u1
  spmm_edges<<<eg, eb, 0, stream>>>(i1, ui_vals, cols, rows, u2);
  spmm_edges<<<eg, eb, 0, stream>>>(u1, iu_vals, rows, cols, i2);
  // u3 = A  @ i2 ; i3 = A^T @ u2
  spmm_edges<<<eg, eb, 0, stream>>>(i2, ui_vals, cols, rows, u3);
  spmm_edges<<<eg, eb, 0, stream>>>(u2, iu_vals, rows, cols, i3);

  combine4<<<((int)UF + 255) / 256, 256, 0, stream>>>(embed_user, u1, u2, u3,
                                                      gcn_u, (int)UF);
  combine4<<<((int)IF + 255) / 256, 256, 0, stream>>>(embed_item, i1, i2, i3,
                                                      gcn_i, (int)IF);

  // 8192 waves per side: (BATCH/16) * 32 threads / 256 per block = 1024 blocks
  int cblocks = (BATCH / 16) * 32 / 256;
  contrastive_wmma<<<cblocks, 256, 0, stream>>>(
      user, item_i, item_j, degree_U, old_U, gcn_u, 1.0f / (float)U_NUM,
      (float*)d_out);
  contrastive_wmma<<<cblocks, 256, 0, stream>>>(
      user_, item_i_, item_j_, degree_I, old_I, gcn_i, 1.0f / (float)I_NUM,
      (float*)d_out);
}